// ModuleCorrelation_48112223649941
// MI455X (gfx1250) — compile-verified
//
#include <hip/hip_runtime.h>

typedef __attribute__((ext_vector_type(2))) float v2f;
typedef __attribute__((ext_vector_type(8))) float v8f;

#define CORR_PAD 4
#define C_TOT 256
#define H_TOT 128
#define W_TOT 128
#define NDISP 81   // 9x9

// Workgroup: 128 threads = 4 waves. Supertile = 8x8 pixels per workgroup.
// Each wave computes a 4x4 sub-tile (M=16) x its 12x12 neighborhood (N=144,
// 9 WMMA tiles) pairwise-dot matrix over C=256 via V_WMMA_F32_16X16X4_F32.
//
// LDS layout is channel-PAIR interleaved: [pair][pixel][2], so a WMMA A/B
// fragment (two consecutive channels for one half-wave) is a single
// contiguous ds_load_b64.
__global__ __launch_bounds__(128) void ModuleCorrelation_wmma_kernel(
    const float* __restrict__ first,
    const float* __restrict__ second,
    float* __restrict__ out) {
  // 9216 floats = 36KB. During compute:
  //   [0..4095]    second staging: 8 ch-pairs x 256 neighborhood pixels x 2
  //   [4096..5119] first  staging: 8 ch-pairs x 64 supertile pixels x 2
  // After compute: accumulator dump (4 waves x 9 tiles x 256 = 9216 floats).
  __shared__ float lds[9216];
  float* lds_s = lds;          // second, interleaved pairs
  float* lds_f = lds + 4096;   // first,  interleaved pairs

  const int tid  = threadIdx.x;
  const int lane = tid & 31;
  const int wave = tid >> 5;

  const int b  = blockIdx.z;
  const int Y0 = blockIdx.y * 8;
  const int X0 = blockIdx.x * 8;

  // sub-tile origin of this wave inside the supertile
  const int sy = (wave >> 1) * 4;
  const int sx = (wave & 1) * 4;

  // A-fragment row (pixel) for this lane: m = lane%16 -> 4x4 sub-tile pixel
  const int m    = lane & 15;
  const int apix = (sy + (m >> 2)) * 8 + (sx + (m & 3));
  const int aoff = apix * 2;          // byte-pair offset within a pair block
  const int hw   = lane >> 4;         // half-wave: holds K rows {4k, 4k+1} or {4k+2, 4k+3}

  // B-fragment pixel positions (columns) for the 9 N-tiles: neighborhood of
  // this sub-tile is 12x12 starting at (sy, sx) in the 16x16 staged block.
  int boff[9];
#pragma unroll
  for (int t = 0; t < 9; ++t) {
    int nn = t * 16 + (lane & 15);
    int r = nn / 12;
    int c = nn - r * 12;
    boff[t] = ((sy + r) * 16 + (sx + c)) * 2;
  }

  v8f acc[9];
#pragma unroll
  for (int t = 0; t < 9; ++t) acc[t] = (v8f){0.f, 0.f, 0.f, 0.f, 0.f, 0.f, 0.f, 0.f};

  const size_t plane = (size_t)H_TOT * W_TOT;
  const float* firstB  = first  + (size_t)b * C_TOT * plane;
  const float* secondB = second + (size_t)b * C_TOT * plane;

  for (int c0 = 0; c0 < C_TOT; c0 += 16) {
    __syncthreads();

    // ---- Stage first: 8 ch-pairs x 64 pixels. Unit = 2 channels x 2 pixels.
    // 256 units, 2 per thread. Always in bounds; float2 loads are 8B aligned
    // (X0 multiple of 8, pixel index even).
#pragma unroll
    for (int it = 0; it < 2; ++it) {
      int u = tid + it * 128;
      int pr = u >> 5;              // channel pair 0..7
      int pix0 = (u & 31) << 1;     // even pixel index 0..62
      int gy = Y0 + (pix0 >> 3), gx = X0 + (pix0 & 7);
      const float* p0 = firstB + (size_t)(c0 + pr * 2) * plane + gy * W_TOT + gx;
      float2 va = *(const float2*)p0;            // channel 2*pr
      float2 vb = *(const float2*)(p0 + plane);  // channel 2*pr+1
      float4 st; st.x = va.x; st.y = vb.x; st.z = va.y; st.w = vb.y;
      *(float4*)&lds_f[pr * 128 + pix0 * 2] = st;
    }

    // ---- Stage second neighborhood (16x16, zero-padded): 8 ch-pairs x 256
    // pixels. Unit = 2 channels x 2 pixels. 1024 units, 8 per thread.
#pragma unroll
    for (int it = 0; it < 8; ++it) {
      int u = tid + it * 128;
      int pr = u >> 7;              // channel pair 0..7
      int pix0 = (u & 127) << 1;    // even pixel 0..254 (both in same 16-wide row)
      int r = pix0 >> 4, cc = pix0 & 15;
      int gy = Y0 + r - CORR_PAD;
      int gx = X0 + cc - CORR_PAD;  // even
      float a0 = 0.f, a1 = 0.f, b0 = 0.f, b1 = 0.f;
      if ((unsigned)gy < (unsigned)H_TOT) {
        const float* p0 = secondB + (size_t)(c0 + pr * 2) * plane + gy * W_TOT;
        const float* p1 = p0 + plane;
        if (gx >= 0 && gx + 1 < W_TOT) {   // interior: aligned float2 loads
          float2 va = *(const float2*)(p0 + gx);
          float2 vb = *(const float2*)(p1 + gx);
          a0 = va.x; a1 = va.y; b0 = vb.x; b1 = vb.y;
        } else {                            // border: per-element
          if ((unsigned)gx < (unsigned)W_TOT)       { a0 = p0[gx];     b0 = p1[gx]; }
          if ((unsigned)(gx + 1) < (unsigned)W_TOT) { a1 = p0[gx + 1]; b1 = p1[gx + 1]; }
        }
      }
      float4 st; st.x = a0; st.y = b0; st.z = a1; st.w = b1;
      *(float4*)&lds_s[pr * 512 + pix0 * 2] = st;
    }

    __syncthreads();

    // ---- 4 k-steps of K=4: batch all fragment loads (b64 each), then 9
    // back-to-back WMMAs so only one dscnt wait per k-step.
#pragma unroll
    for (int kk = 0; kk < 4; ++kk) {
      const int pr = kk * 2 + hw;   // channel pair held by this half-wave
      v2f a = *(const v2f*)&lds_f[pr * 128 + aoff];
      v2f bb[9];
#pragma unroll
      for (int t = 0; t < 9; ++t)
        bb[t] = *(const v2f*)&lds_s[pr * 512 + boff[t]];
#pragma unroll
      for (int t = 0; t < 9; ++t)
        acc[t] = __builtin_amdgcn_wmma_f32_16x16x4_f32(
            /*neg_a=*/false, a, /*neg_b=*/false, bb[t],
            /*c_mod=*/(short)0, acc[t], /*reuse_a=*/false, /*reuse_b=*/false);
    }
  }

  __syncthreads();
  // Dump D matrices to LDS: lds[wave*2304 + t*256 + row*16 + col]
  {
    const int ncol  = lane & 15;
    const int mbase = (lane >> 4) ? 8 : 0;
#pragma unroll
    for (int t = 0; t < 9; ++t)
#pragma unroll
      for (int v = 0; v < 8; ++v)
        lds[wave * 2304 + t * 256 + (mbase + v) * 16 + ncol] = acc[t][v];
  }
  __syncthreads();

  // Gather banded entries and write out[b][d][y][x], scaled by 1/C (mean).
  const float scale = 1.0f / (float)C_TOT;
  for (int i = tid; i < NDISP * 64; i += 128) {
    const int d = i >> 6;          // 0..80
    const int p = i & 63;          // supertile pixel
    const int py = p >> 3, px = p & 7;
    const int dy = d / 9 - CORR_PAD;
    const int dx = d % 9 - CORR_PAD;
    const int w  = ((py >> 2) << 1) | (px >> 2);
    const int ly = py & 3, lx = px & 3;
    const int mm = ly * 4 + lx;
    const int r  = ly + dy + CORR_PAD;   // 0..11
    const int c  = lx + dx + CORR_PAD;   // 0..11
    const int nn = r * 12 + c;
    const float v = lds[w * 2304 + (nn >> 4) * 256 + mm * 16 + (nn & 15)];
    out[(((size_t)b * NDISP + d) * H_TOT + (Y0 + py)) * W_TOT + (X0 + px)] = v * scale;
  }
}

extern "C" void kernel_launch(void* const* d_in, const int* in_sizes, int n_in,
                              void* d_out, int out_size, void* d_ws, size_t ws_size,
                              hipStream_t stream) {
  (void)in_sizes; (void)n_in; (void)out_size; (void)d_ws; (void)ws_size;
  const float* first  = (const float*)d_in[0];
  const float* second = (const float*)d_in[1];
  float* out = (float*)d_out;
  dim3 grid(W_TOT / 8, H_TOT / 8, 4);
  ModuleCorrelation_wmma_kernel<<<grid, 128, 0, stream>>>(first, second, out);
}